// SecondOrderMixtureRNN_40656160424381
// MI455X (gfx1250) — compile-verified
//
#include <hip/hip_runtime.h>
#include <hip/hip_bf16.h>
#include <math.h>

// Problem sizes
#define BB   16
#define TT   256
#define DD   256
#define MM   32
#define VV   50257
#define LSTR 260   // padded LDS row stride (floats): row*260 % 64 = row*4 -> 16 distinct banks
#define MSTR 260   // stride for staged mover K/V rows

typedef float v2f __attribute__((ext_vector_type(2)));
typedef float v8f __attribute__((ext_vector_type(8)));

__device__ __forceinline__ v8f wmma_f32_k4(v2f a, v2f b, v8f c) {
  // D = A(16x4,f32) * B(4x16,f32) + C(16x16,f32)
  return __builtin_amdgcn_wmma_f32_16x16x4_f32(false, a, false, b, (short)0, c, false, false);
}

__device__ __forceinline__ float wave_sum32(float v) {
#pragma unroll
  for (int m = 16; m >= 1; m >>= 1) v += __shfl_xor(v, m, 32);
  return v;
}

// ---------------------------------------------------------------------------
// Kernel 0: precompute mover-token K/V projections (step-invariant):
//   mvK1 = mover1 @ Wk1.T + bk1   (32x256)
//   mvV1 = mover1 @ Wv1.T + bv1
//   mvK2 = mover2 @ Wk2.T + bk2
//   mvV2 = mover2 @ Wv2.T + bv2
// ws layout (floats): [0)=mvK1, [8192)=mvV1, [16384)=mvK2, [24576)=mvV2
// ---------------------------------------------------------------------------
__global__ __launch_bounds__(256) void pre_kernel(
    const float* __restrict__ mover1, const float* __restrict__ mover2,
    const float* __restrict__ Wk1, const float* __restrict__ bk1,
    const float* __restrict__ Wv1, const float* __restrict__ bv1,
    const float* __restrict__ Wk2, const float* __restrict__ bk2,
    const float* __restrict__ Wv2, const float* __restrict__ bv2,
    float* __restrict__ ws) {
  int id = blockIdx.x * blockDim.x + threadIdx.x;  // 4*32*256 = 32768 tasks
  if (id >= 4 * MM * DD) return;
  int mat  = id >> 13;         // 0..3
  int m    = (id >> 8) & 31;   // mover row
  int dcol = id & 255;         // output feature
  const float* mv = (mat < 2) ? mover1 : mover2;
  const float* W;
  const float* bb;
  switch (mat) {
    case 0:  W = Wk1; bb = bk1; break;
    case 1:  W = Wv1; bb = bv1; break;
    case 2:  W = Wk2; bb = bk2; break;
    default: W = Wv2; bb = bv2; break;
  }
  float acc = bb[dcol];
  const float* mr = mv + m * DD;
  const float* wr = W + (size_t)dcol * DD;
#pragma unroll 8
  for (int k = 0; k < DD; ++k) acc = fmaf(mr[k], wr[k], acc);
  ws[id] = acc;
}

// ---------------------------------------------------------------------------
// Kernel 1: persistent single-workgroup recurrent scan (512 threads = 16 waves)
// Wave w owns output-column tile [16w, 16w+16) and batch row b == w for the
// lane-parallel phases. All 6 per-step matvecs use V_WMMA_F32_16X16X4_F32.
// Mover K/V rows staged in LDS; embedding gather software-pipelined.
// ---------------------------------------------------------------------------
__global__ __launch_bounds__(512) void scan_kernel(
    const float* __restrict__ hid, const int* __restrict__ seq,
    const float* __restrict__ emb,
    const float* __restrict__ Wq1, const float* __restrict__ bq1,
    const float* __restrict__ Wk1, const float* __restrict__ bk1,
    const float* __restrict__ Wv1, const float* __restrict__ bv1,
    const float* __restrict__ Wq2, const float* __restrict__ bq2,
    const float* __restrict__ Wk2, const float* __restrict__ bk2,
    const float* __restrict__ Wv2, const float* __restrict__ bv2,
    const float* __restrict__ gamma, const float* __restrict__ beta,
    const float* __restrict__ mvK1, const float* __restrict__ mvV1,
    const float* __restrict__ mvK2, const float* __restrict__ mvV2,
    float* __restrict__ out) {
  __shared__ float sZ0[BB * LSTR];
  __shared__ float sZ1[BB * LSTR];
  __shared__ float sQ1[BB * LSTR];
  __shared__ float sK1[BB * LSTR];
  __shared__ float sV1[BB * LSTR];
  __shared__ float sQ2[BB * LSTR];
  __shared__ float sK2[BB * LSTR];
  __shared__ float sV2[BB * LSTR];
  __shared__ float sMK1[MM * MSTR];   // staged mover projections (step-invariant)
  __shared__ float sMV1[MM * MSTR];
  __shared__ float sMK2[MM * MSTR];
  __shared__ float sMV2[MM * MSTR];
  __shared__ float sSc[2 * BB * 33];  // attention scores -> probs (in place)

  const int tid  = threadIdx.x;
  const int lane = tid & 31;
  const int wave = tid >> 5;   // 0..15  (batch row / N-tile owner)
  const int half = lane >> 4;  // 0|1
  const int l15  = lane & 15;
  const int n0   = wave * 16;  // this wave's output-column tile
  const int ncol = n0 + l15;   // per-lane output column / weight row
  const int c0   = lane * 8;   // per-lane column span for row-parallel phases

  // Hoisted step-invariant per-lane values
  const float* wq1r = Wq1 + (size_t)ncol * DD;
  const float* wk1r = Wk1 + (size_t)ncol * DD;
  const float* wv1r = Wv1 + (size_t)ncol * DD;
  const float* wq2r = Wq2 + (size_t)ncol * DD;
  const float* wk2r = Wk2 + (size_t)ncol * DD;
  const float* wv2r = Wv2 + (size_t)ncol * DD;
  const float bq1v = bq1[ncol], bk1v = bk1[ncol], bv1v = bv1[ncol];
  const float bq2v = bq2[ncol], bk2v = bk2[ncol], bv2v = bv2[ncol];
  float gv[8], bv[8];
#pragma unroll
  for (int i = 0; i < 8; ++i) { gv[i] = gamma[c0 + i]; bv[i] = beta[c0 + i]; }

  // Stage mover K/V projections into LDS (read every step on the serial path)
  for (int i = tid; i < MM * DD; i += 512) {
    const int m = i >> 8, c = i & 255;
    sMK1[m * MSTR + c] = mvK1[i];
    sMV1[m * MSTR + c] = mvV1[i];
    sMK2[m * MSTR + c] = mvK2[i];
    sMV2[m * MSTR + c] = mvV2[i];
  }

  // Init carry from hidden_state (B,2,D)
  {
    const int b = wave;
#pragma unroll
    for (int i = 0; i < 8; ++i) {
      sZ0[b * LSTR + c0 + i] = hid[b * 2 * DD + c0 + i];
      sZ1[b * LSTR + c0 + i] = hid[b * 2 * DD + DD + c0 + i];
    }
  }

  // Software pipeline: preload x_0
  float xcur[8];
  {
    const int tok = seq[wave * TT + 0];
    const float* er = emb + (size_t)tok * DD + c0;
#pragma unroll
    for (int i = 0; i < 8; ++i) xcur[i] = 16.0f * er[i];
  }
  __syncthreads();

  for (int t = 0; t < TT; ++t) {
    // --- Phase 1: z0 += x_t (registers, wave-local row) --------------------
    {
      const int b = wave;
#pragma unroll
      for (int i = 0; i < 8; ++i) sZ0[b * LSTR + c0 + i] += xcur[i];
    }
    __syncthreads();

    // Issue next step's embedding loads now; latency hidden under phases 2-4.
    float xnext[8];
    {
      const int tn  = (t + 1 < TT) ? (t + 1) : (TT - 1);
      const int tok = seq[wave * TT + tn];
      const float* er = emb + (size_t)tok * DD + c0;
#pragma unroll
      for (int i = 0; i < 8; ++i) xnext[i] = 16.0f * er[i];
    }

    // --- Phase 2: Q1/K1/V1 from z0, Q2/K2/V2 from z1 via fp32 WMMA ---------
    {
      v8f aq1 = {}, ak1 = {}, av1 = {}, aq2 = {}, ak2 = {}, av2 = {};
#pragma unroll 8
      for (int k0 = 0; k0 < DD; k0 += 4) {
        const int kk = k0 + 2 * half;
        v2f a0; a0.x = sZ0[l15 * LSTR + kk]; a0.y = sZ0[l15 * LSTR + kk + 1];
        v2f a1; a1.x = sZ1[l15 * LSTR + kk]; a1.y = sZ1[l15 * LSTR + kk + 1];
        v2f b_;
        b_.x = wq1r[kk]; b_.y = wq1r[kk + 1]; aq1 = wmma_f32_k4(a0, b_, aq1);
        b_.x = wk1r[kk]; b_.y = wk1r[kk + 1]; ak1 = wmma_f32_k4(a0, b_, ak1);
        b_.x = wv1r[kk]; b_.y = wv1r[kk + 1]; av1 = wmma_f32_k4(a0, b_, av1);
        b_.x = wq2r[kk]; b_.y = wq2r[kk + 1]; aq2 = wmma_f32_k4(a1, b_, aq2);
        b_.x = wk2r[kk]; b_.y = wk2r[kk + 1]; ak2 = wmma_f32_k4(a1, b_, ak2);
        b_.x = wv2r[kk]; b_.y = wv2r[kk + 1]; av2 = wmma_f32_k4(a1, b_, av2);
      }
#pragma unroll
      for (int r = 0; r < 8; ++r) {
        const int m = r + 8 * half;  // batch row (C/D layout)
        sQ1[m * LSTR + ncol] = aq1[r] + bq1v;
        sK1[m * LSTR + ncol] = ak1[r] + bk1v;
        sV1[m * LSTR + ncol] = av1[r] + bv1v;
        sQ2[m * LSTR + ncol] = aq2[r] + bq2v;
        sK2[m * LSTR + ncol] = ak2[r] + bk2v;
        sV2[m * LSTR + ncol] = av2[r] + bv2v;
      }
    }
    __syncthreads();

    // --- Phase 3a: attention scores (2 attns x 16 batches x 33 keys) -------
    for (int idx = wave; idx < 2 * BB * 33; idx += 16) {
      const int a   = idx / (BB * 33);
      const int rem = idx % (BB * 33);
      const int b   = rem / 33;
      const int j   = rem % 33;
      const float* qrow = (a == 0 ? sQ1 : sQ2) + b * LSTR;
      const float* krow =
          (j == 0) ? ((a == 0 ? sK1 : sK2) + b * LSTR)
                   : ((a == 0 ? sMK1 : sMK2) + (j - 1) * MSTR);
      float acc = 0.f;
#pragma unroll
      for (int i = 0; i < 8; ++i) acc = fmaf(qrow[c0 + i], krow[c0 + i], acc);
      acc = wave_sum32(acc);
      if (lane == 0) sSc[idx] = acc * (1.0f / 16.0f);  // / sqrt(D)
    }
    __syncthreads();

    // --- Phase 3b: softmax over 33 keys per (attn, batch) ------------------
    if (tid < 2 * BB) {
      float* row = sSc + tid * 33;
      float mx = row[0];
      for (int j = 1; j < 33; ++j) mx = fmaxf(mx, row[j]);
      float s = 0.f;
      for (int j = 0; j < 33; ++j) { float e = expf(row[j] - mx); row[j] = e; s += e; }
      const float inv = 1.0f / s;
      for (int j = 0; j < 33; ++j) row[j] *= inv;
    }
    __syncthreads();

    // --- Phase 3c+4 (fused, wave-local): dg1+dg2+residual, layernorm -------
    {
      const int b = wave;
      const float* p1 = sSc + b * 33;
      const float* p2 = sSc + (BB + b) * 33;
      float accv[8];
#pragma unroll
      for (int i = 0; i < 8; ++i)
        accv[i] = sZ0[b * LSTR + c0 + i]
                + p1[0] * sV1[b * LSTR + c0 + i]
                + p2[0] * sV2[b * LSTR + c0 + i];
      for (int j = 1; j < 33; ++j) {
        const float* v1r = sMV1 + (j - 1) * MSTR;
        const float* v2r = sMV2 + (j - 1) * MSTR;
        const float w1 = p1[j], w2 = p2[j];
#pragma unroll
        for (int i = 0; i < 8; ++i)
          accv[i] = fmaf(w1, v1r[c0 + i], fmaf(w2, v2r[c0 + i], accv[i]));
      }
      // layernorm directly on registers (no LDS round trip)
      float s = 0.f, s2 = 0.f;
#pragma unroll
      for (int i = 0; i < 8; ++i) { s += accv[i]; s2 = fmaf(accv[i], accv[i], s2); }
      s  = wave_sum32(s);
      s2 = wave_sum32(s2);
      const float mu  = s * (1.0f / DD);
      const float var = s2 * (1.0f / DD) - mu * mu;
      const float inv = rsqrtf(var + 1e-6f);
#pragma unroll
      for (int i = 0; i < 8; ++i) {
        const float y = (accv[i] - mu) * inv * gv[i] + bv[i];
        sZ0[b * LSTR + c0 + i] = y;
        sZ1[b * LSTR + c0 + i] = y;
      }
    }
    // No barrier: next phase that reads cross-wave state (phase 2) is already
    // preceded by the phase-1 barrier; phases 1/3c/4 touch only row b==wave.

#pragma unroll
    for (int i = 0; i < 8; ++i) xcur[i] = xnext[i];
  }

  // Emit concat([z0, z1], axis=1) -> out[0 : B*2*D)  (wave-local rows)
  {
    const int b = wave;
#pragma unroll
    for (int i = 0; i < 8; ++i) {
      const float z = sZ0[b * LSTR + c0 + i];
      out[b * 2 * DD + c0 + i]      = z;
      out[b * 2 * DD + DD + c0 + i] = z;
    }
  }
}

// ---------------------------------------------------------------------------
// Kernel 2: logits = z0 @ Wvoc.T + bvoc, WMMA-tiled over V (memory-bound:
// 51.5MB of Wvoc @ 23.3TB/s). Writes straight into the y region of d_out.
// ---------------------------------------------------------------------------
__global__ __launch_bounds__(256) void logits_kernel(
    const float* __restrict__ zsrc /* = d_out, z at b*2*D + c */,
    const float* __restrict__ Wvoc, const float* __restrict__ bvoc,
    float* __restrict__ lg /* = d_out + 8192, [16][50257] */) {
  __shared__ float sZ[BB * LSTR];
  const int tid  = threadIdx.x;
  const int lane = tid & 31;
  const int wave = tid >> 5;   // 8 waves per block, one V-tile each
  const int half = lane >> 4;
  const int l15  = lane & 15;

  for (int i = tid; i < BB * DD; i += 256) {
    const int b = i >> 8, c = i & 255;
    sZ[b * LSTR + c] = zsrc[b * 2 * DD + c];
  }
  __syncthreads();

  const int v0 = (blockIdx.x * 8 + wave) * 16;
  if (v0 >= VV) return;  // wave-uniform -> EXEC stays all-ones for WMMA

  const int vr = v0 + l15;
  const int vc = (vr < VV) ? vr : (VV - 1);  // clamp OOB rows (stores guarded)
  const float* wrow = Wvoc + (size_t)vc * DD;

  v8f acc = {};
#pragma unroll 8
  for (int k0 = 0; k0 < DD; k0 += 4) {
    const int kk = k0 + 2 * half;
    v2f a; a.x = sZ[l15 * LSTR + kk]; a.y = sZ[l15 * LSTR + kk + 1];
    v2f b; b.x = wrow[kk];            b.y = wrow[kk + 1];
    acc = wmma_f32_k4(a, b, acc);
  }
#pragma unroll
  for (int r = 0; r < 8; ++r) {
    const int m = r + 8 * half;  // batch
    const int n = v0 + l15;      // vocab id
    if (n < VV) lg[(size_t)m * VV + n] = acc[r] + bvoc[n];
  }
}

// ---------------------------------------------------------------------------
// Kernel 3: in-place log_softmax over V per batch row.
// ---------------------------------------------------------------------------
__global__ __launch_bounds__(256) void lsm_kernel(float* __restrict__ y) {
  __shared__ float red[256];
  const int b = blockIdx.x, tid = threadIdx.x;
  float* row = y + (size_t)b * VV;

  float mx = -3.402823466e+38f;
  for (int i = tid; i < VV; i += 256) mx = fmaxf(mx, row[i]);
  red[tid] = mx;
  __syncthreads();
  for (int s = 128; s > 0; s >>= 1) {
    if (tid < s) red[tid] = fmaxf(red[tid], red[tid + s]);
    __syncthreads();
  }
  mx = red[0];
  __syncthreads();

  float sum = 0.f;
  for (int i = tid; i < VV; i += 256) sum += expf(row[i] - mx);
  red[tid] = sum;
  __syncthreads();
  for (int s = 128; s > 0; s >>= 1) {
    if (tid < s) red[tid] += red[tid + s];
    __syncthreads();
  }
  const float lse = mx + logf(red[0]);
  __syncthreads();

  for (int i = tid; i < VV; i += 256) row[i] = row[i] - lse;
}

// ---------------------------------------------------------------------------
extern "C" void kernel_launch(void* const* d_in, const int* in_sizes, int n_in,
                              void* d_out, int out_size, void* d_ws, size_t ws_size,
                              hipStream_t stream) {
  const float* hid    = (const float*)d_in[0];
  const int*   seq    = (const int*)d_in[1];
  const float* emb    = (const float*)d_in[2];
  const float* Wq1    = (const float*)d_in[3];
  const float* bq1    = (const float*)d_in[4];
  const float* Wk1    = (const float*)d_in[5];
  const float* bk1    = (const float*)d_in[6];
  const float* Wv1    = (const float*)d_in[7];
  const float* bv1    = (const float*)d_in[8];
  const float* Wq2    = (const float*)d_in[9];
  const float* bq2    = (const float*)d_in[10];
  const float* Wk2    = (const float*)d_in[11];
  const float* bk2    = (const float*)d_in[12];
  const float* Wv2    = (const float*)d_in[13];
  const float* bv2    = (const float*)d_in[14];
  const float* mover1 = (const float*)d_in[15];
  const float* mover2 = (const float*)d_in[16];
  const float* gamma  = (const float*)d_in[17];
  const float* beta   = (const float*)d_in[18];
  const float* Wvoc   = (const float*)d_in[19];
  const float* bvoc   = (const float*)d_in[20];

  float* ws  = (float*)d_ws;   // [0)=mvK1 [8192)=mvV1 [16384)=mvK2 [24576)=mvV2
  float* out = (float*)d_out;  // [0:8192)=concat(z0,z1); [8192:)=y (16x50257)

  // 0) step-invariant mover K/V projections
  pre_kernel<<<128, 256, 0, stream>>>(mover1, mover2, Wk1, bk1, Wv1, bv1,
                                      Wk2, bk2, Wv2, bv2, ws);

  // 1) persistent recurrent scan (single workgroup, 16 waves)
  scan_kernel<<<1, 512, 0, stream>>>(hid, seq, emb,
                                     Wq1, bq1, Wk1, bk1, Wv1, bv1,
                                     Wq2, bq2, Wk2, bk2, Wv2, bv2,
                                     gamma, beta,
                                     ws, ws + 8192, ws + 16384, ws + 24576,
                                     out);

  // 2) vocab projection -> logits directly into y region
  const int vtiles = (VV + 15) / 16;          // 3142
  const int blocks = (vtiles + 7) / 8;        // 393
  logits_kernel<<<blocks, 256, 0, stream>>>(out, Wvoc, bvoc, out + BB * 2 * DD);

  // 3) in-place log_softmax per batch row
  lsm_kernel<<<BB, 256, 0, stream>>>(out + BB * 2 * DD);
}